// SlotAttention_87110526697914
// MI455X (gfx1250) — compile-verified
//
#include <hip/hip_runtime.h>
#include <math.h>

#define NNODES 100000
#define NPART  5000
#define NEDGES 1000000
#define DNODE  128
#define DATT   20

typedef float v2f __attribute__((ext_vector_type(2)));
typedef float v8f __attribute__((ext_vector_type(8)));

__device__ __forceinline__ v8f wmma_f32(v2f a, v2f b, v8f c) {
    // V_WMMA_F32_16X16X4_F32 : D = A(16x4) * B(4x16) + C(16x16), all fp32 (exact)
    return __builtin_amdgcn_wmma_f32_16x16x4_f32(false, a, false, b, (short)0, c, false, false);
}

__device__ __forceinline__ float sigmoidf_(float x) { return 1.0f / (1.0f + expf(-x)); }

// 16x16 tile GEMM over K=128: A staged in LDS [16][132], B row-major in memory
// (ldb = row stride), no branches in the loop.
__device__ __forceinline__ v8f gemm_tile_B(const float (*A)[132],
                                           const float* __restrict__ B, int ldb,
                                           int n, int half, int lm) {
    v8f acc = {0.f,0.f,0.f,0.f,0.f,0.f,0.f,0.f};
    #pragma unroll 8
    for (int kk = 0; kk < 32; ++kk) {
        const int k = kk * 4 + 2 * half;
        v2f a; a.x = A[lm][k]; a.y = A[lm][k + 1];
        v2f b; b.x = B[k * ldb + n]; b.y = B[(k + 1) * ldb + n];
        acc = wmma_f32(a, b, acc);
    }
    return acc;
}

// Same but B = W^T where W is [N][128] row-major: contiguous float2 per lane.
__device__ __forceinline__ v8f gemm_tile_BT(const float (*A)[132],
                                            const float* __restrict__ W,
                                            int n, int half, int lm) {
    v8f acc = {0.f,0.f,0.f,0.f,0.f,0.f,0.f,0.f};
    #pragma unroll 8
    for (int kk = 0; kk < 32; ++kk) {
        const int k = kk * 4 + 2 * half;
        v2f a; a.x = A[lm][k]; a.y = A[lm][k + 1];
        const float2 bl = *(const float2*)&W[(size_t)n * 128 + k];
        v2f b; b.x = bl.x; b.y = bl.y;
        acc = wmma_f32(a, b, acc);
    }
    return acc;
}

// -------------------------------------------------------------------------
// Kernel 1: fused keys = X@key_W+key_b and vals = X@val_W+val_b.
// block = 128 threads (4 waves), one 16-row M-tile per block.
// key_W staged zero-padded to [128][32] in LDS so key tiles are branch-free.
// -------------------------------------------------------------------------
__global__ __launch_bounds__(128)
void k_keys_vals(const float* __restrict__ X,
                 const float* __restrict__ keyW, const float* __restrict__ keyB,
                 const float* __restrict__ valW, const float* __restrict__ valB,
                 float* __restrict__ keys, float* __restrict__ vals) {
    __shared__ float As[16][132];       // 16x128 A tile (padded stride)
    __shared__ float Bk[128][32];       // key_W padded with zeros (cols 20..31)
    const int m0  = blockIdx.x * 16;
    const int tid = threadIdx.x;

    #pragma unroll
    for (int i = 0; i < 4; ++i) {
        int idx = tid + i * 128;
        int row = idx >> 5, c4 = idx & 31;
        *(float4*)&As[row][c4 * 4] = *(const float4*)&X[(size_t)(m0 + row) * DNODE + c4 * 4];
    }
    #pragma unroll
    for (int i = 0; i < 32; ++i) {
        int idx = tid + i * 128;        // 4096 entries
        int r = idx >> 5, c = idx & 31;
        Bk[r][c] = (c < DATT) ? keyW[r * DATT + c] : 0.0f;
    }
    __syncthreads();

    const int wave = tid >> 5, lane = tid & 31;
    const int half = lane >> 4, lm = lane & 15;

    // --- key tiles (waves 0 and 1; n-tile = wave) ---
    if (wave < 2) {
        const int n = wave * 16 + lm;
        v8f acc = gemm_tile_B(As, &Bk[0][0], 32, n, half, lm);
        if (n < DATT) {
            const float bias = keyB[n];
            #pragma unroll
            for (int r = 0; r < 8; ++r)
                keys[(size_t)(m0 + r + 8 * half) * DATT + n] = acc[r] + bias;
        }
    }

    // --- val tiles (8 tiles, 2 per wave) ---
    for (int tv = wave; tv < 8; tv += 4) {
        const int n = tv * 16 + lm;
        v8f acc = gemm_tile_B(As, valW, DNODE, n, half, lm);
        const float bias = valB[n];
        #pragma unroll
        for (int r = 0; r < 8; ++r)
            vals[(size_t)(m0 + r + 8 * half) * DNODE + n] = acc[r] + bias;
    }
}

// -------------------------------------------------------------------------
// Kernel 2: query[p][a] = concat(ph,gr)[p] @ q_W + q_b   (tiny)
// -------------------------------------------------------------------------
__global__ __launch_bounds__(256)
void k_query(const float* __restrict__ ph, const float* __restrict__ gr,
             const float* __restrict__ qW, const float* __restrict__ qb,
             float* __restrict__ query) {
    int idx = blockIdx.x * blockDim.x + threadIdx.x;
    if (idx >= NPART * DATT) return;
    int p = idx / DATT, a = idx % DATT;
    float acc = qb[a];
    const float* phr = ph + (size_t)p * DNODE;
    const float* grr = gr + (size_t)p * DNODE;
    for (int k = 0; k < DNODE; ++k) acc += phr[k] * qW[k * DATT + a];
    for (int k = 0; k < DNODE; ++k) acc += grr[k] * qW[(DNODE + k) * DATT + a];
    query[idx] = acc;
}

// -------------------------------------------------------------------------
// Kernel 3: zero the segment-sum accumulator
// -------------------------------------------------------------------------
__global__ __launch_bounds__(256)
void k_zero(float* __restrict__ p, int n) {
    int i = blockIdx.x * blockDim.x + threadIdx.x;
    if (i < n) p[i] = 0.0f;
}

// -------------------------------------------------------------------------
// Kernel 4: wave-per-edge attention + scatter-add (all L2-resident).
// att = dot(keys[src], query[dst]) / sqrt(20);  wsum[dst] += att * vals[src]
// -------------------------------------------------------------------------
__global__ __launch_bounds__(256)
void k_edges(const float* __restrict__ keys, const float* __restrict__ query,
             const float* __restrict__ vals,
             const int* __restrict__ src, const int* __restrict__ dst,
             float* __restrict__ wsum) {
    const int lane = threadIdx.x & 31;
    const int e = blockIdx.x * 8 + (threadIdx.x >> 5);
    if (e >= NEDGES) return;
    const int s = src[e];
    const int d = dst[e];
    float v = 0.0f;
    if (lane < DATT) v = keys[(size_t)s * DATT + lane] * query[(size_t)d * DATT + lane];
    #pragma unroll
    for (int off = 16; off > 0; off >>= 1) v += __shfl_xor(v, off, 32);
    const float att = v * 0.223606797749979f;           // 1/sqrt(20)
    const float4 vv = *(const float4*)&vals[(size_t)s * DNODE + lane * 4];
    float* wrow = wsum + (size_t)d * DNODE + lane * 4;
    atomicAdd(wrow + 0, att * vv.x);
    atomicAdd(wrow + 1, att * vv.y);
    atomicAdd(wrow + 2, att * vv.z);
    atomicAdd(wrow + 3, att * vv.w);
}

// -------------------------------------------------------------------------
// Kernel 5: fused GRU (WMMA gemms) + LayerNorm + MLP + residual.
// block = 128 threads, 16 particles per block.
// -------------------------------------------------------------------------
__global__ __launch_bounds__(128)
void k_tail(const float* __restrict__ wsum, const float* __restrict__ ph,
            const float* __restrict__ Wih, const float* __restrict__ Whh,
            const float* __restrict__ bih, const float* __restrict__ bhh,
            const float* __restrict__ lng, const float* __restrict__ lnb,
            const float* __restrict__ W1, const float* __restrict__ b1,
            const float* __restrict__ W2, const float* __restrict__ b2,
            float* __restrict__ out) {
    __shared__ float Aw[16][132];      // wsum tile -> later h_new
    __shared__ float Ah[16][132];      // ph tile   -> later layernorm out
    __shared__ float Gi[16][388];      // gi = wsum @ Wih^T + bih
    __shared__ float Gh[16][388];      // gh = ph   @ Whh^T + bhh
    __shared__ float H1[16][68];       // mlp hidden

    const int p0  = blockIdx.x * 16;
    const int tid = threadIdx.x;

    #pragma unroll
    for (int i = 0; i < 4; ++i) {
        int idx = tid + i * 128;
        int row = idx >> 5, c4 = idx & 31;
        int pr = p0 + row; if (pr >= NPART) pr = NPART - 1;
        *(float4*)&Aw[row][c4 * 4] = *(const float4*)&wsum[(size_t)pr * DNODE + c4 * 4];
        *(float4*)&Ah[row][c4 * 4] = *(const float4*)&ph  [(size_t)pr * DNODE + c4 * 4];
    }
    __syncthreads();

    const int wave = tid >> 5, lane = tid & 31;
    const int half = lane >> 4, lm = lane & 15;

    // --- gi = wsum @ Wih^T + bih : 24 uniform n-tiles of 16 ---
    for (int t = wave; t < 24; t += 4) {
        const int n = t * 16 + lm;
        v8f acc = gemm_tile_BT(Aw, Wih, n, half, lm);
        const float bias = bih[n];
        #pragma unroll
        for (int r = 0; r < 8; ++r)
            Gi[r + 8 * half][n] = acc[r] + bias;
    }
    // --- gh = ph @ Whh^T + bhh ---
    for (int t = wave; t < 24; t += 4) {
        const int n = t * 16 + lm;
        v8f acc = gemm_tile_BT(Ah, Whh, n, half, lm);
        const float bias = bhh[n];
        #pragma unroll
        for (int r = 0; r < 8; ++r)
            Gh[r + 8 * half][n] = acc[r] + bias;
    }
    __syncthreads();

    // --- GRU gates: h_new into Aw ---
    {
        const int c = tid;                               // 0..127
        for (int m = 0; m < 16; ++m) {
            float r = sigmoidf_(Gi[m][c]       + Gh[m][c]);
            float z = sigmoidf_(Gi[m][128 + c] + Gh[m][128 + c]);
            float n = tanhf    (Gi[m][256 + c] + r * Gh[m][256 + c]);
            Aw[m][c] = (1.0f - z) * n + z * Ah[m][c];
        }
    }
    __syncthreads();

    // --- LayerNorm per row (wave handles 4 rows), result into Ah ---
    for (int m = wave * 4; m < wave * 4 + 4; ++m) {
        float s = 0.0f;
        #pragma unroll
        for (int i = 0; i < 4; ++i) s += Aw[m][lane + 32 * i];
        #pragma unroll
        for (int off = 16; off > 0; off >>= 1) s += __shfl_xor(s, off, 32);
        const float mu = s * (1.0f / 128.0f);
        float q = 0.0f;
        #pragma unroll
        for (int i = 0; i < 4; ++i) { float d = Aw[m][lane + 32 * i] - mu; q += d * d; }
        #pragma unroll
        for (int off = 16; off > 0; off >>= 1) q += __shfl_xor(q, off, 32);
        const float rstd = rsqrtf(q * (1.0f / 128.0f) + 1e-5f);
        #pragma unroll
        for (int i = 0; i < 4; ++i) {
            int c = lane + 32 * i;
            Ah[m][c] = (Aw[m][c] - mu) * rstd * lng[c] + lnb[c];
        }
    }
    __syncthreads();

    // --- MLP layer 1: 128 -> 64, ReLU ---
    {
        const int j = tid & 63, hh = tid >> 6;
        for (int m = hh * 8; m < hh * 8 + 8; ++m) {
            float acc = b1[j];
            for (int k = 0; k < DNODE; ++k) acc += Ah[m][k] * W1[k * 64 + j];
            H1[m][j] = fmaxf(acc, 0.0f);
        }
    }
    __syncthreads();

    // --- MLP layer 2: 64 -> 128, + residual ph, store ---
    {
        const int c = tid;
        for (int m = 0; m < 16; ++m) {
            const int p = p0 + m;
            if (p >= NPART) break;
            float acc = b2[c];
            for (int k = 0; k < 64; ++k) acc += H1[m][k] * W2[k * DNODE + c];
            out[(size_t)p * DNODE + c] = ph[(size_t)p * DNODE + c] + acc;
        }
    }
}

// -------------------------------------------------------------------------
extern "C" void kernel_launch(void* const* d_in, const int* in_sizes, int n_in,
                              void* d_out, int out_size, void* d_ws, size_t ws_size,
                              hipStream_t stream) {
    const float* nodes_hidden = (const float*)d_in[0];
    const float* part_hidden  = (const float*)d_in[1];
    const float* global_rep   = (const float*)d_in[2];
    const int*   src          = (const int*)d_in[3];
    const int*   dst          = (const int*)d_in[4];
    const float* key_W = (const float*)d_in[5];
    const float* key_b = (const float*)d_in[6];
    const float* val_W = (const float*)d_in[7];
    const float* val_b = (const float*)d_in[8];
    const float* q_W   = (const float*)d_in[9];
    const float* q_b   = (const float*)d_in[10];
    const float* W_ih  = (const float*)d_in[11];
    const float* W_hh  = (const float*)d_in[12];
    const float* b_ih  = (const float*)d_in[13];
    const float* b_hh  = (const float*)d_in[14];
    const float* ln_g  = (const float*)d_in[15];
    const float* ln_b  = (const float*)d_in[16];
    const float* mlp_W1 = (const float*)d_in[17];
    const float* mlp_b1 = (const float*)d_in[18];
    const float* mlp_W2 = (const float*)d_in[19];
    const float* mlp_b2 = (const float*)d_in[20];
    float* outp = (float*)d_out;

    // workspace layout (fp32): keys | vals | query | wsum  (~62.2 MB)
    float* keys  = (float*)d_ws;
    float* vals  = keys  + (size_t)NNODES * DATT;
    float* query = vals  + (size_t)NNODES * DNODE;
    float* wsum  = query + (size_t)NPART * DATT;

    k_keys_vals<<<NNODES / 16, 128, 0, stream>>>(nodes_hidden, key_W, key_b,
                                                 val_W, val_b, keys, vals);
    k_query<<<(NPART * DATT + 255) / 256, 256, 0, stream>>>(part_hidden, global_rep,
                                                            q_W, q_b, query);
    k_zero<<<(NPART * DNODE + 255) / 256, 256, 0, stream>>>(wsum, NPART * DNODE);
    k_edges<<<NEDGES / 8, 256, 0, stream>>>(keys, query, vals, src, dst, wsum);
    k_tail<<<(NPART + 15) / 16, 128, 0, stream>>>(wsum, part_hidden,
                                                  W_ih, W_hh, b_ih, b_hh,
                                                  ln_g, ln_b, mlp_W1, mlp_b1,
                                                  mlp_W2, mlp_b2, outp);
}